// FillNet_60172491817331
// MI455X (gfx1250) — compile-verified
//
#include <hip/hip_runtime.h>
#include <hip/hip_bf16.h>

typedef __attribute__((ext_vector_type(2))) float v2f;
typedef __attribute__((ext_vector_type(8))) float v8f;

#define M_NODES            1024
#define WAVES_PER_BLOCK    8
#define BLOCK_THREADS      (WAVES_PER_BLOCK * 32)
#define QUERIES_PER_WAVE   16
#define QUERIES_PER_BLOCK  (WAVES_PER_BLOCK * QUERIES_PER_WAVE)  // 128

// out[q][c] = sum_m W2[c][m] * K(q,m) / sum_m K(q,m),
//   K(q,m) = exp(-||X_q - P_m||^2 / (2*sigmaSq_m)) = exp2(D * scale_m)
//
// Reduction over m with V_WMMA_F32_16X16X4_F32 (transposed orientation):
//   D(16x16) += A(16x4) x B(4x16)
//   A rows = weight channels {R,G,B,den=1} (rows 4..15 are duplicates -> dead D rows)
//   B cols = 16 queries; B[k][q] = K(node m0+k, query q), computed per lane
// Only D rows 0..3 (acc VGPRs 0..3, lanes 0..15) are read out.
__global__ __launch_bounds__(BLOCK_THREADS)
void rbf_wmma_kernel(const int* __restrict__ X,
                     const int* __restrict__ P,
                     const float* __restrict__ W2,
                     const float* __restrict__ sigmaSq,
                     float* __restrict__ out,
                     int nQuery)
{
    __shared__ float sNode[M_NODES * 4];        // 16 KB: x, y, scale, 0 per node
    __shared__ float sWp[(M_NODES / 2) * 8];    // 16 KB: pair-interleaved weights

    const int tid = threadIdx.x;

    // Cooperative stage into LDS (32 KB total).
    for (int i = tid; i < M_NODES; i += BLOCK_THREADS) {
        float px = (float)P[2 * i];
        float py = (float)P[2 * i + 1];
        float sc = -0.7213475204444817f / sigmaSq[i];   // -log2(e)/2 / sigmaSq
        sNode[4 * i + 0] = px;
        sNode[4 * i + 1] = py;
        sNode[4 * i + 2] = sc;
        sNode[4 * i + 3] = 0.0f;
        // pair-interleaved: [Wr(m) Wr(m+1) | Wg(m) Wg(m+1) | Wb(m) Wb(m+1) | 1 1]
        int pair = i >> 1, j = i & 1;
        sWp[pair * 8 + 0 + j] = W2[0 * M_NODES + i];
        sWp[pair * 8 + 2 + j] = W2[1 * M_NODES + i];
        sWp[pair * 8 + 4 + j] = W2[2 * M_NODES + i];
        sWp[pair * 8 + 6 + j] = 1.0f;                    // denominator row
    }
    __syncthreads();

    const int lane = tid & 31;
    const int wave = tid >> 5;
    const int half = lane >> 4;       // which K-pair: k={0,1} vs k={2,3}
    const int sub  = lane & 15;       // B column (query in tile) / A row

    int q = blockIdx.x * QUERIES_PER_BLOCK + wave * QUERIES_PER_WAVE + sub;
    int qload = q < nQuery ? q : (nQuery - 1);   // keep EXEC all-ones for WMMA
    const float qx = (float)X[2 * qload];
    const float qy = (float)X[2 * qload + 1];

    const float4* nodes4 = (const float4*)sNode;
    // Loop-invariant lane offsets (all lanes load unconditionally).
    const int nOff = half * 2;                       // node index offset
    const int wOff = half * 8 + (sub & 3) * 2;       // float offset into pair record

    v8f acc = {};

    #pragma unroll 8
    for (int m0 = 0; m0 < M_NODES; m0 += 4) {
        float4 n0 = nodes4[m0 + nOff];               // ds_load_b96/b128, broadcast
        float4 n1 = nodes4[m0 + nOff + 1];
        v2f a = *(const v2f*)&sWp[4 * m0 + wOff];    // one aligned ds_load_b64

        float dx0 = qx - n0.x, dy0 = qy - n0.y;
        float dx1 = qx - n1.x, dy1 = qy - n1.y;
        float k0 = __builtin_amdgcn_exp2f(__builtin_fmaf(dx0, dx0, dy0 * dy0) * n0.z);
        float k1 = __builtin_amdgcn_exp2f(__builtin_fmaf(dx1, dx1, dy1 * dy1) * n1.z);

        v2f b; b.x = k0; b.y = k1;

        // (neg_a, A, neg_b, B, c_mod, C, reuse_a, reuse_b)
        acc = __builtin_amdgcn_wmma_f32_16x16x4_f32(
            false, a, false, b, (short)0, acc, false, false);
    }

    // D layout: VGPR r, lanes 0-15 -> row r. Rows 0..2 = RGB, row 3 = den.
    if (lane < 16 && q < nQuery) {
        float inv = 1.0f / acc[3];
        float* o = out + 3 * q;
        o[0] = acc[0] * inv;
        o[1] = acc[1] * inv;
        o[2] = acc[2] * inv;
    }
}

extern "C" void kernel_launch(void* const* d_in, const int* in_sizes, int n_in,
                              void* d_out, int out_size, void* d_ws, size_t ws_size,
                              hipStream_t stream) {
    const int*   X     = (const int*)d_in[0];
    const int*   P     = (const int*)d_in[1];
    const float* W2    = (const float*)d_in[2];
    const float* sigma = (const float*)d_in[3];
    float*       out   = (float*)d_out;

    int nQuery = in_sizes[0] / 2;
    int grid = (nQuery + QUERIES_PER_BLOCK - 1) / QUERIES_PER_BLOCK;

    rbf_wmma_kernel<<<grid, BLOCK_THREADS, 0, stream>>>(X, P, W2, sigma, out, nQuery);
}